// NRBS_22617297780916
// MI455X (gfx1250) — compile-verified
//
#include <hip/hip_runtime.h>
#include <math.h>

// ---------------------------------------------------------------------------
// NRBS fused pipeline for MI455X (gfx1250, wave32).
//   N=100000 points, n=32 bases, m=16 clusters, k=32 neighbours, MU=100.
// Memory-bound problem (~65 MB streaming + L2-resident gather); fp32 kept
// end-to-end; the final encoded@convolved_basis contraction runs on the
// matrix unit via V_WMMA_F32_16X16X4_F32.
// ---------------------------------------------------------------------------

#define NPTS 100000
#define NBAS 32      // n
#define MCLU 16      // m
#define KNEI 32      // k
#define MUF  100.0f

typedef __attribute__((ext_vector_type(2))) float v2f;
typedef __attribute__((ext_vector_type(8))) float v8f;

// ---- workspace layout (float offsets) -------------------------------------
#define WS_ENCPART 0                       // [32][32] GEMV partials
#define WS_ENCODED (WS_ENCPART + 1024)     // [32]
#define WS_INVW2T  (WS_ENCODED + 32)       // [16][32]  inv_w2 transposed [c][i]
#define WS_RDENT   (WS_INVW2T + 512)       // [32][32]  1/denom transposed [kk][i]
#define DEN_BLKS   224
#define WS_DENPART (WS_RDENT + 1024)       // [DEN_BLKS][32][32]
#define WS_DECT    (WS_DENPART + DEN_BLKS*1024)  // [N][32] transposed decoder

// ===========================================================================
// K1a: encoded partials.  encoded[i] = sum_j W[i,j]*x[j].
// grid = (32 chunks, 32 rows), block = 256. Deterministic tree reduce.
// ===========================================================================
__global__ void k_enc_partial(const float* __restrict__ x,
                              const float* __restrict__ W,
                              float* __restrict__ part) {
  const int row = blockIdx.y, chunk = blockIdx.x;
  const int CS = (NPTS + 31) / 32;              // 3125
  const int j0 = chunk * CS;
  const int j1 = (j0 + CS < NPTS) ? (j0 + CS) : NPTS;
  float s = 0.f;
  for (int j = j0 + threadIdx.x; j < j1; j += blockDim.x)
    s += x[j] * W[row * NPTS + j];
  __shared__ float red[256];
  red[threadIdx.x] = s;
  __syncthreads();
  for (int o = 128; o > 0; o >>= 1) {
    if (threadIdx.x < o) red[threadIdx.x] += red[threadIdx.x + o];
    __syncthreads();
  }
  if (threadIdx.x == 0) part[row * 32 + chunk] = red[0];
}

// ===========================================================================
// K1b: finish encoded, then bandwidths -> inv_w2 transposed [c][i].
// 1 block, 512 threads.
// ===========================================================================
__global__ void k_enc_finish(const float* __restrict__ part,
                             const float* __restrict__ enc_b,
                             const float* __restrict__ bw_w,
                             const float* __restrict__ bw_b,
                             float* __restrict__ encoded,
                             float* __restrict__ invw2t) {
  __shared__ float enc_s[NBAS];
  const int t = threadIdx.x;
  if (t < NBAS) {
    float s = 0.f;
    for (int c = 0; c < 32; ++c) s += part[t * 32 + c];
    s += enc_b[t];
    enc_s[t] = s;
    encoded[t] = s;
  }
  __syncthreads();
  // row r = i*MCLU + c of bw_w (reshape(n,m) row-major)
  if (t < NBAS * MCLU) {
    float s = 0.f;
    for (int q = 0; q < NBAS; ++q) s += enc_s[q] * bw_w[t * NBAS + q];
    s += bw_b[t];
    float band = (1.f / (1.f + expf(-s))) / 60.f;
    float wm = band * MUF;
    const int i = t / MCLU, c = t % MCLU;
    invw2t[c * NBAS + i] = 1.f / (wm * wm);
  }
}

// ===========================================================================
// K_trans: decoder [32][N] -> dec_t [N][32] so a neighbour gather is one
// contiguous 128B row (L2-resident). LDS tiled transpose, block (32,32).
// ===========================================================================
__global__ void k_dec_transpose(const float* __restrict__ dec,
                                float* __restrict__ dect) {
  __shared__ float tile[32][33];
  const int jb = blockIdx.x * 32;
  const int tx = threadIdx.x, ty = threadIdx.y;
  tile[ty][tx] = dec[ty * NPTS + jb + tx];     // coalesced read of row ty
  __syncthreads();
  dect[(jb + ty) * 32 + tx] = tile[tx][ty];    // coalesced write
}

// ===========================================================================
// K2: denom partials. denom[i][kk] = sum_j relu(1 - d2[j,kk]*invw2[i,lab[j]]).
// lane = kk (coalesced 128B d-row per point), 32 register accumulators per
// lane over i. Per-block deterministic LDS combine -> global partials.
// grid = DEN_BLKS, block = 256 (8 waves).
// ===========================================================================
__global__ void k_denom_partial(const float* __restrict__ d,
                                const int* __restrict__ lab,
                                const float* __restrict__ invw2t,
                                float* __restrict__ part) {
  __shared__ float iw[MCLU * NBAS];
  __shared__ float red[8 * 1024];
  for (int t = threadIdx.x; t < MCLU * NBAS; t += blockDim.x) iw[t] = invw2t[t];
  __syncthreads();
  const int lane = threadIdx.x & 31, wv = threadIdx.x >> 5;
  const int gw = blockIdx.x * 8 + wv, nw = gridDim.x * 8;
  float acc[NBAS];
#pragma unroll
  for (int i = 0; i < NBAS; ++i) acc[i] = 0.f;
  for (int j = gw; j < NPTS; j += nw) {
    const int c = lab[j];                       // wave-uniform
    const float dv = d[j * KNEI + lane];        // coalesced 128B
    const float d2 = dv * dv;
#pragma unroll
    for (int i = 0; i < NBAS; ++i)
      acc[i] += fmaxf(1.f - d2 * iw[c * NBAS + i], 0.f);
  }
#pragma unroll
  for (int i = 0; i < NBAS; ++i) red[wv * 1024 + i * 32 + lane] = acc[i];
  __syncthreads();
  for (int e = threadIdx.x; e < 1024; e += blockDim.x) {
    float s = 0.f;
#pragma unroll
    for (int w = 0; w < 8; ++w) s += red[w * 1024 + e];
    part[blockIdx.x * 1024 + e] = s;            // layout [i*32+kk]
  }
}

// ===========================================================================
// K2b: reduce block partials, emit 1/denom transposed [kk][i].
// 1 block, 1024 threads, fixed-order sum (deterministic).
// ===========================================================================
__global__ void k_denom_finish(const float* __restrict__ part, int nblk,
                               float* __restrict__ rdent) {
  const int t = threadIdx.x;                    // t = i*32+kk
  float s = 0.f;
  for (int b = 0; b < nblk; ++b) s += part[b * 1024 + t];
  const int i = t >> 5, kk = t & 31;
  rdent[kk * 32 + i] = 1.f / s;
}

// ===========================================================================
// K3: fused gather + normalize + output.
//   G[i,col] = sum_kk (1/denom[i,kk]) * relu(1 - d2*invw2[i,c]) * dec_t[nid][i]
//   out[16-pt tile] = encoded[1x32] @ G[32x16]  via 8x V_WMMA_F32_16X16X4_F32
// lane = i: dec_t gather is one coalesced 128B row per (j,kk) with a scalar
// base (readlane'd nid). G staged in stride-17 LDS (conflict-free), intra-wave
// LDS RAW fenced with s_wait_dscnt. EXEC is full at every WMMA.
// grid = 200, block = 256 (8 waves), wave-stride over 6250 tiles.
// ===========================================================================
__global__ void k_fused_out(const float* __restrict__ d,
                            const int* __restrict__ nid,
                            const int* __restrict__ lab,
                            const float* __restrict__ dect,
                            const float* __restrict__ invw2t,
                            const float* __restrict__ rdent,
                            const float* __restrict__ encoded,
                            float* __restrict__ out) {
  __shared__ float iw[MCLU * NBAS];
  __shared__ float rd[NBAS * NBAS];
  __shared__ float enc_s[NBAS];
  __shared__ float g[8][32 * 17];               // per-wave G, pad 17 (gcd(17,64)=1)
  for (int t = threadIdx.x; t < MCLU * NBAS; t += blockDim.x) iw[t] = invw2t[t];
  for (int t = threadIdx.x; t < NBAS * NBAS; t += blockDim.x) rd[t] = rdent[t];
  if (threadIdx.x < NBAS) enc_s[threadIdx.x] = encoded[threadIdx.x];
  __syncthreads();

  const int lane = threadIdx.x & 31, wv = threadIdx.x >> 5;
  const int gw = blockIdx.x * 8 + wv, nw = gridDim.x * 8;
  float* gbuf = g[wv];
  const int NT = NPTS / 16;                     // 6250, exact

  for (int tile = gw; tile < NT; tile += nw) {
    const int j0 = tile * 16;
    for (int p = 0; p < 16; ++p) {
      const int j = j0 + p;
      const int c = lab[j];                     // wave-uniform
      const float s = iw[c * NBAS + lane];      // invw2[c][i=lane], conflict-free
      const int nl = nid[j * KNEI + lane];      // coalesced 128B
      const float dv = d[j * KNEI + lane];      // coalesced 128B
      const float d2 = dv * dv;
      float acc = 0.f;
#pragma unroll
      for (int kk = 0; kk < KNEI; ++kk) {
        const int nb = __shfl(nl, kk, 32);      // v_readlane -> SGPR base
        const float d2k = __shfl(d2, kk, 32);
        const float win = fmaxf(1.f - d2k * s, 0.f);
        const float dec = dect[nb * 32 + lane]; // coalesced 128B gather (L2)
        acc += rd[kk * 32 + lane] * win * dec;  // rd conflict-free
      }
      gbuf[lane * 17 + p] = acc;                // G[i=lane][col=p]
    }
    // intra-wave LDS RAW: writes above, cross-lane reads below (DS in-order)
    asm volatile("s_wait_dscnt 0" ::: "memory");

    // out[col] = sum_i encoded[i] * G[i][col]; A rows all = encoded chunk.
    v8f cacc = {0.f, 0.f, 0.f, 0.f, 0.f, 0.f, 0.f, 0.f};
    const int half = (lane >> 4) & 1;           // K=0/1 (lanes 0-15), K=2/3 (16-31)
    const int col = lane & 15;
#pragma unroll
    for (int s8 = 0; s8 < 8; ++s8) {
      const int kb = s8 * 4;
      v2f a, b;
      a.x = enc_s[kb + 2 * half];               // A[m, k] = encoded[kb+k]
      a.y = enc_s[kb + 2 * half + 1];
      b.x = gbuf[(kb + 2 * half) * 17 + col];   // B[k, col] = G[kb+k][col]
      b.y = gbuf[(kb + 2 * half + 1) * 17 + col];
      cacc = __builtin_amdgcn_wmma_f32_16x16x4_f32(
          false, a, false, b, (short)0, cacc, false, false);
    }
    if (lane < 16) out[j0 + lane] = cacc[0];    // D row M=0
  }
}

// ===========================================================================
extern "C" void kernel_launch(void* const* d_in, const int* in_sizes, int n_in,
                              void* d_out, int out_size, void* d_ws, size_t ws_size,
                              hipStream_t stream) {
  const float* x     = (const float*)d_in[0];
  const float* enc_w = (const float*)d_in[1];
  const float* enc_b = (const float*)d_in[2];
  const float* dec   = (const float*)d_in[3];
  const float* bw_w  = (const float*)d_in[4];
  const float* bw_b  = (const float*)d_in[5];
  const float* nd    = (const float*)d_in[6];
  const int*   nid   = (const int*)d_in[7];
  const int*   lab   = (const int*)d_in[8];
  float* out = (float*)d_out;
  float* ws  = (float*)d_ws;

  float* enc_part = ws + WS_ENCPART;
  float* encoded  = ws + WS_ENCODED;
  float* invw2t   = ws + WS_INVW2T;
  float* rdent    = ws + WS_RDENT;
  float* denpart  = ws + WS_DENPART;
  float* dect     = ws + WS_DECT;

  // 1) encoded = enc_w @ x (two-stage deterministic GEMV)
  k_enc_partial<<<dim3(32, 32), 256, 0, stream>>>(x, enc_w, enc_part);
  k_enc_finish<<<1, 512, 0, stream>>>(enc_part, enc_b, bw_w, bw_b, encoded, invw2t);
  // 2) decoder transpose for contiguous gathers
  k_dec_transpose<<<NPTS / 32, dim3(32, 32), 0, stream>>>(dec, dect);
  // 3) window denominators (sum over N axis)
  k_denom_partial<<<DEN_BLKS, 256, 0, stream>>>(nd, lab, invw2t, denpart);
  k_denom_finish<<<1, 1024, 0, stream>>>(denpart, DEN_BLKS, rdent);
  // 4) fused gather + normalize + WMMA contraction -> out
  k_fused_out<<<200, 256, 0, stream>>>(nd, nid, lab, dect, invw2t, rdent,
                                       encoded, out);
}